// MultiHeadAttention_36490042147145
// MI455X (gfx1250) — compile-verified
//
#include <hip/hip_runtime.h>

typedef __attribute__((ext_vector_type(2))) float v2f;
typedef __attribute__((ext_vector_type(8))) float v8f;

#define D_MODEL 1024
#define NHEAD   16
#define DK      64
#define SEQ     2048
#define BATCH   2
#define MTOT    (BATCH * SEQ)   // 4096
#define NT      8               // N-tiles (of 16) per wave in the GEMM

static __device__ __forceinline__ v8f wmma4(v2f a, v2f b, v8f c) {
  // D(16x16,f32) = A(16x4,f32) x B(4x16,f32) + C   -- full fp32 tensor op
  return __builtin_amdgcn_wmma_f32_16x16x4_f32(
      /*neg_a=*/false, a, /*neg_b=*/false, b,
      /*c_mod=*/(short)0, c, /*reuse_a=*/false, /*reuse_b=*/false);
}

// ---------------------------------------------------------------------------
// Trapezoidal quadrature weights: w[s] = 0.5*(|c[s]-c[s-1]| + |c[s+1]-c[s]|),
// one-sided at the ends. Collapses the reference's trapezoid softmax into a
// standard per-key-weighted softmax.
// ---------------------------------------------------------------------------
__global__ void quadw_kernel(const float* __restrict__ coords,
                             float* __restrict__ w) {
  int s = blockIdx.x * blockDim.x + threadIdx.x;
  if (s >= SEQ) return;
  float left  = (s > 0)       ? fabsf(coords[s] - coords[s - 1]) : 0.0f;
  float right = (s < SEQ - 1) ? fabsf(coords[s + 1] - coords[s]) : 0.0f;
  w[s] = 0.5f * (left + right);
}

// ---------------------------------------------------------------------------
// C[m,n] = sum_k X[m,k] * W[n,k] + bias[n]
// X: [MTOT, 1024] row-major, W: [1024, 1024] row-major (torch-style [out,in]).
// blockIdx.z selects one of up to 3 (W, bias, out) sets (fused Q/K/V).
// out_mode 0: row-major [MTOT,1024]; out_mode 1: per-head [b,h,s,d] layout.
// Each wave computes a 16(M) x 128(N) strip with 8 f32 WMMA accumulators,
// software-pipelined: fragments for k+4 are prefetched into registers while
// the 8 WMMAs for k execute, so s_wait_loadcnt only drains the older group.
// ---------------------------------------------------------------------------
__global__ void gemm_xwt_kernel(const float* __restrict__ X,
                                const float* __restrict__ W0,
                                const float* __restrict__ B0,
                                float* __restrict__ O0,
                                const float* __restrict__ W1,
                                const float* __restrict__ B1,
                                float* __restrict__ O1,
                                const float* __restrict__ W2,
                                const float* __restrict__ B2,
                                float* __restrict__ O2,
                                int out_mode) {
  const float* W = W0; const float* bias = B0; float* out = O0;
  if (blockIdx.z == 1) { W = W1; bias = B1; out = O1; }
  else if (blockIdx.z == 2) { W = W2; bias = B2; out = O2; }

  const int lane = threadIdx.x & 31;
  const int wave = threadIdx.x >> 5;
  const int row  = lane & 15;   // fragment row (A) / col (B,D)
  const int hi   = lane >> 4;   // half-wave select
  const int m0   = blockIdx.x * 128 + wave * 16;
  const int n0   = blockIdx.y * (16 * NT);

  const float* xrow = X + (size_t)(m0 + row) * D_MODEL + 2 * hi;
  const float* wr[NT];
#pragma unroll
  for (int t = 0; t < NT; ++t)
    wr[t] = W + (size_t)(n0 + t * 16 + row) * D_MODEL + 2 * hi;

  v8f acc[NT];
#pragma unroll
  for (int t = 0; t < NT; ++t) acc[t] = {};

  // ---- software-pipelined K loop (double-buffered fragments) ----
  v2f ac = *(const v2f*)(xrow);
  v2f bc[NT];
#pragma unroll
  for (int t = 0; t < NT; ++t) bc[t] = *(const v2f*)(wr[t]);

#pragma unroll 2
  for (int k0 = 0; k0 < D_MODEL; k0 += 4) {
    const int kn = (k0 + 4) & (D_MODEL - 1);  // wrap: last prefetch re-reads k=0
    // prefetch next k-chunk fragments (independent of current WMMAs)
    v2f an = *(const v2f*)(xrow + kn);
    v2f bn[NT];
#pragma unroll
    for (int t = 0; t < NT; ++t) bn[t] = *(const v2f*)(wr[t] + kn);
    // compute on current fragments
#pragma unroll
    for (int t = 0; t < NT; ++t) acc[t] = wmma4(ac, bc[t], acc[t]);
    // rotate buffers
    ac = an;
#pragma unroll
    for (int t = 0; t < NT; ++t) bc[t] = bn[t];
  }

#pragma unroll
  for (int c = 0; c < NT; ++c) {
    const int n  = n0 + c * 16 + row;
    const float bv = bias[n];
#pragma unroll
    for (int r = 0; r < 8; ++r) {
      const int m = m0 + r + 8 * hi;
      const float val = acc[c][r] + bv;
      size_t off;
      if (out_mode == 0) {
        off = (size_t)m * D_MODEL + n;
      } else {
        const int b = m >> 11, s = m & (SEQ - 1);
        const int h = n >> 6,  d = n & (DK - 1);
        off = (((size_t)(b * NHEAD + h)) * SEQ + s) * DK + d;
      }
      out[off] = val;
    }
  }
}

// ---------------------------------------------------------------------------
// Flash attention with per-key quadrature weights, per (b,h) head.
// Each wave owns 16 query rows; loops over 128 key tiles of 16.
// Scores via f32 WMMA (K=64 -> 16 chained WMMAs), online max/denom with
// 16-lane shuffle reductions, P transposed D-layout -> A-layout through
// padded LDS, then P x V via 16 more WMMAs into 16x64 accumulators.
// Output written in [b,s,h,d] ( = [m, 1024] row-major) for the final GEMM.
// ---------------------------------------------------------------------------
__global__ void attn_kernel(const float* __restrict__ Q,
                            const float* __restrict__ K,
                            const float* __restrict__ V,
                            const float* __restrict__ w,
                            float* __restrict__ O) {
  __shared__ __align__(16) float lds[8][16][18];  // pad 18 -> conflict-free

  const int lane = threadIdx.x & 31;
  const int wave = threadIdx.x >> 5;
  const int row  = lane & 15;
  const int hi   = lane >> 4;
  const int bh   = blockIdx.y;                 // b*NHEAD + h
  const int q0   = blockIdx.x * 128 + wave * 16;

  const float* Qh = Q + (size_t)bh * SEQ * DK;
  const float* Kh = K + (size_t)bh * SEQ * DK;
  const float* Vh = V + (size_t)bh * SEQ * DK;

  // Preload the wave's Q block as 16 A-fragments (loop-invariant).
  v2f qf[16];
  {
    const float* qrow = Qh + (size_t)(q0 + row) * DK + 2 * hi;
#pragma unroll
    for (int c = 0; c < 16; ++c) qf[c] = *(const v2f*)(qrow + 4 * c);
  }

  v8f acc[4];
  acc[0] = {}; acc[1] = {}; acc[2] = {}; acc[3] = {};
  float rmax[8], rden[8];
#pragma unroll
  for (int r = 0; r < 8; ++r) { rmax[r] = -1.0e30f; rden[r] = 0.0f; }

  const float scale = 0.125f;  // 1/sqrt(DK)

  for (int t = 0; t < SEQ / 16; ++t) {
    const int s0 = t * 16;

    // ---- scores tile: S(16x16) = Qblk(16x64) x K(s0:s0+16,:)^T ----
    // 16 independent fragment loads against a serial WMMA chain: the
    // scheduler can issue all loads up front and drain loadcnt gradually.
    v8f sf = {};
    {
      const float* krow = Kh + (size_t)(s0 + row) * DK + 2 * hi;
      v2f kf[16];
#pragma unroll
      for (int c = 0; c < 16; ++c) kf[c] = *(const v2f*)(krow + 4 * c);
#pragma unroll
      for (int c = 0; c < 16; ++c) sf = wmma4(qf[c], kf[c], sf);
    }

    const float wv = w[s0 + row];  // quadrature weight of key (s0 + col)

    // ---- online weighted softmax update ----
    float p[8];
#pragma unroll
    for (int r = 0; r < 8; ++r) {
      float sc = sf[r] * scale;
      float tm = sc;
#pragma unroll
      for (int o = 1; o < 16; o <<= 1) tm = fmaxf(tm, __shfl_xor(tm, o, 32));
      const float nm = fmaxf(rmax[r], tm);
      const float cf = __expf(rmax[r] - nm);
      rmax[r] = nm;
      const float e = __expf(sc - nm) * wv;
      float rs = e;
#pragma unroll
      for (int o = 1; o < 16; o <<= 1) rs += __shfl_xor(rs, o, 32);
      rden[r] = rden[r] * cf + rs;
      p[r] = e;
#pragma unroll
      for (int c = 0; c < 4; ++c) acc[c][r] *= cf;
    }

    // ---- transpose P: D-layout -> A-layout via LDS ----
    __syncthreads();  // WAR vs previous iteration's loads
#pragma unroll
    for (int r = 0; r < 8; ++r) lds[wave][r + 8 * hi][row] = p[r];
    __syncthreads();  // RAW
    v2f pa[4];
#pragma unroll
    for (int c = 0; c < 4; ++c)
      pa[c] = *(const v2f*)&lds[wave][row][4 * c + 2 * hi];

    // ---- acc(16x64) += P(16x16) x V(s0:s0+16, 0:64) ----
#pragma unroll
    for (int c = 0; c < 4; ++c) {
      const float* vr = Vh + (size_t)(s0 + 4 * c + 2 * hi) * DK + row;
#pragma unroll
      for (int nt = 0; nt < 4; ++nt) {
        v2f bf = { vr[nt * 16], vr[DK + nt * 16] };
        acc[nt] = wmma4(pa[c], bf, acc[nt]);
      }
    }
  }

  // ---- epilogue: divide by denom, write [b,s,h,d] ----
  const int b = bh >> 4, h = bh & 15;
#pragma unroll
  for (int nt = 0; nt < 4; ++nt) {
#pragma unroll
    for (int r = 0; r < 8; ++r) {
      const int s = q0 + r + 8 * hi;
      const int d = nt * 16 + row;
      O[((size_t)(b * SEQ + s)) * D_MODEL + h * DK + d] = acc[nt][r] / rden[r];
    }
  }
}

// ---------------------------------------------------------------------------
extern "C" void kernel_launch(void* const* d_in, const int* in_sizes, int n_in,
                              void* d_out, int out_size, void* d_ws, size_t ws_size,
                              hipStream_t stream) {
  const float* x      = (const float*)d_in[0];
  const float* coords = (const float*)d_in[1];
  const float* Wq     = (const float*)d_in[2];
  const float* bq     = (const float*)d_in[3];
  const float* Wk     = (const float*)d_in[4];
  const float* bk     = (const float*)d_in[5];
  const float* Wv     = (const float*)d_in[6];
  const float* bv     = (const float*)d_in[7];
  const float* Wo     = (const float*)d_in[8];
  const float* bo     = (const float*)d_in[9];
  float* out = (float*)d_out;

  float* ws = (float*)d_ws;
  const size_t MK = (size_t)MTOT * D_MODEL;  // 4096*1024 floats
  float* q_ws = ws;
  float* k_ws = ws + MK;
  float* v_ws = ws + 2 * MK;
  float* o_ws = ws + 3 * MK;
  float* w_ws = ws + 4 * MK;

  // 1) quadrature weights
  quadw_kernel<<<dim3((SEQ + 255) / 256), dim3(256), 0, stream>>>(coords, w_ws);

  // 2) fused Q/K/V projections (z selects matrix), per-head output layout
  gemm_xwt_kernel<<<dim3(MTOT / 128, D_MODEL / (16 * NT), 3), dim3(256), 0, stream>>>(
      x, Wq, bq, q_ws, Wk, bk, k_ws, Wv, bv, v_ws, /*out_mode=*/1);

  // 3) weighted flash attention
  attn_kernel<<<dim3(SEQ / 128, BATCH * NHEAD), dim3(256), 0, stream>>>(
      q_ws, k_ws, v_ws, w_ws, o_ws);

  // 4) output projection straight into d_out
  gemm_xwt_kernel<<<dim3(MTOT / 128, D_MODEL / (16 * NT), 1), dim3(256), 0, stream>>>(
      o_ws, Wo, bo, out, Wo, bo, out, Wo, bo, out, /*out_mode=*/0);
}